// CombinedSurvivalModel_32289564131465
// MI455X (gfx1250) — compile-verified
//
#include <hip/hip_runtime.h>
#include <hip/hip_bf16.h>
#include <math.h>

#define B_SZ   128
#define IN_DIM 400000
#define D0     4096
#define D1     1024
#define D2     512
#define D3     128
#define FANIN  128
#define N_CLIN 8
#define N_BINS 30
#define LN_EPS 1e-5f

// LDS row stride for staged k-panels: 36 floats = 144B -> 16B-aligned rows for
// B128 async writes AND conflict-free ds_load_b64 (gcd(36,64)=4, 16 lanes hit
// 16 distinct banks).
#define KB     32
#define LROW   36

typedef __attribute__((ext_vector_type(2))) float v2f;
typedef __attribute__((ext_vector_type(4))) float v4f;
typedef __attribute__((ext_vector_type(8))) float v8f;
typedef int v4i __attribute__((vector_size(16)));   // matches builtin's pointee type

#if defined(__AMDGCN__) && __has_builtin(__builtin_amdgcn_global_load_async_to_lds_b128) && __has_builtin(__builtin_amdgcn_s_wait_asynccnt)
#define USE_ASYNC_LDS 1
#else
#define USE_ASYNC_LDS 0
#endif

__device__ __forceinline__ void async_copy16(const float* g, float* l) {
#if USE_ASYNC_LDS
  __builtin_amdgcn_global_load_async_to_lds_b128(
      (__attribute__((address_space(1))) v4i*)(g),
      (__attribute__((address_space(3))) v4i*)(l), 0, 0);
#else
  *(v4f*)l = *(const v4f*)g;
#endif
}

__device__ __forceinline__ void async_wait_all() {
#if USE_ASYNC_LDS
  __builtin_amdgcn_s_wait_asynccnt(0);
#endif
}

// ---------------------------------------------------------------------------
// 1) Tiled transpose: x_main[128][400000] -> xT[400000][128]
// ---------------------------------------------------------------------------
__global__ void transpose_kernel(const float* __restrict__ x, float* __restrict__ xT) {
  __shared__ float tile[32][33];
  const int c0 = blockIdx.x * 32;
  const int r0 = blockIdx.y * 32;
  const int tx = threadIdx.x;
  const int ty = threadIdx.y;
#pragma unroll
  for (int i = 0; i < 32; i += 8) {
    int c = c0 + tx;
    int r = r0 + ty + i;
    tile[ty + i][tx] = (c < IN_DIM) ? x[(size_t)r * IN_DIM + c] : 0.f;
  }
  __syncthreads();
#pragma unroll
  for (int i = 0; i < 32; i += 8) {
    int c = c0 + ty + i;
    int r = r0 + tx;
    if (c < IN_DIM) xT[(size_t)c * B_SZ + r] = tile[tx][ty + i];
  }
}

// ---------------------------------------------------------------------------
// 2) Sparse layer 1: block per output o, thread = batch row b. Indices are
//    wave-uniform (scalar loads); xT column reads are contiguous 512B.
// ---------------------------------------------------------------------------
__global__ void gather_l1_kernel(const float* __restrict__ xT,
                                 const int*   __restrict__ cols,
                                 const float* __restrict__ vals,
                                 const float* __restrict__ bias,
                                 float* __restrict__ h) {
  const int o = blockIdx.x;
  const int b = threadIdx.x;             // 128 threads
  const int*   cr = cols + (size_t)o * FANIN;
  const float* vr = vals + (size_t)o * FANIN;
  float s = 0.f;
#pragma unroll 4
  for (int k = 0; k < FANIN; ++k) {
    if (k + 16 < FANIN)
      __builtin_prefetch(xT + (size_t)cr[k + 16] * B_SZ + b, 0, 0);
    s += xT[(size_t)cr[k] * B_SZ + b] * vr[k];
  }
  s += bias[o];
  s = (s > 0.f) ? s : expm1f(s);
  h[(size_t)b * D0 + o] = s;
}

// ---------------------------------------------------------------------------
// Block reduction helper (256 threads = 8 waves, wave32 shuffles).
// ---------------------------------------------------------------------------
__device__ __forceinline__ float block_reduce_sum(float v, float* sm) {
#pragma unroll
  for (int off = 16; off > 0; off >>= 1) v += __shfl_down(v, off, 32);
  const int lane = threadIdx.x & 31;
  const int w    = threadIdx.x >> 5;
  if (lane == 0) sm[w] = v;
  __syncthreads();
  if (w == 0) {
    float r = (lane < 8) ? sm[lane] : 0.f;
#pragma unroll
    for (int off = 4; off > 0; off >>= 1) r += __shfl_down(r, off, 32);
    if (lane == 0) sm[8] = r;
  }
  __syncthreads();
  float r = sm[8];
  __syncthreads();
  return r;
}

// Plain LayerNorm (layer 1 path; ELU already applied by gather kernel).
__global__ void layernorm_kernel(const float* __restrict__ h,
                                 const float* __restrict__ g,
                                 const float* __restrict__ beta,
                                 float* __restrict__ out, int N) {
  __shared__ float sm[9];
  const int b = blockIdx.x;
  const float* row = h + (size_t)b * N;
  float s = 0.f;
  for (int i = threadIdx.x; i < N; i += 256) s += row[i];
  const float mean = block_reduce_sum(s, sm) / (float)N;
  float v = 0.f;
  for (int i = threadIdx.x; i < N; i += 256) {
    float d = row[i] - mean;
    v += d * d;
  }
  const float var = block_reduce_sum(v, sm) / (float)N;
  const float rs  = rsqrtf(var + LN_EPS);
  float* orow = out + (size_t)b * N;
  for (int i = threadIdx.x; i < N; i += 256)
    orow[i] = (row[i] - mean) * rs * g[i] + beta[i];
}

// ---------------------------------------------------------------------------
// Split-K WMMA GEMM with async double-buffered LDS staging.
//   part[s][128][N] = A[128][k-chunk s] @ W[N][k-chunk s]^T
// Block = 256 thr = 8 waves; block owns a 64-wide N slice and one K chunk.
// Per KB=32 k-panel: stage A(128xKB) + W(64xKB) into LDS via ASYNC-to-LDS
// B128 transfers, overlap with WMMA on the other buffer.
//   A 16x4 : lane%16 = M, lane/16 selects K pair (v2f).
//   B 4x16 : lane%16 = N, lane/16 selects K pair (v2f).
//   C/D    : 8 VGPRs; element r -> (M = r + 8*(lane/16), N = lane%16).
// ---------------------------------------------------------------------------
__device__ __forceinline__ void stage_tiles(const float* __restrict__ A,
                                            const float* __restrict__ W,
                                            float* __restrict__ lA,
                                            float* __restrict__ lB,
                                            int nBase, int K, int kw, int tid) {
#pragma unroll
  for (int i = 0; i < 4; ++i) {        // A: 128 rows x 8 chunks of 4 floats
    int c = tid + i * 256;             // 0..1023
    int m = c >> 3, kk = (c & 7) * 4;
    async_copy16(A + (size_t)m * K + kw + kk, lA + m * LROW + kk);
  }
#pragma unroll
  for (int i = 0; i < 2; ++i) {        // W: 64 rows x 8 chunks of 4 floats
    int c = tid + i * 256;             // 0..511
    int n = c >> 3, kk = (c & 7) * 4;
    async_copy16(W + (size_t)(nBase + n) * K + kw + kk, lB + n * LROW + kk);
  }
}

__global__ void gemm_wmma_splitk_kernel(const float* __restrict__ A,
                                        const float* __restrict__ W,
                                        float* __restrict__ part,
                                        int K, int N, int chunk) {
  __shared__ float lA[2][B_SZ * LROW];   // 2 x 18KB
  __shared__ float lB[2][64 * LROW];     // 2 x 9KB
  const int tid   = threadIdx.x;
  const int wave  = tid >> 5;
  const int lane  = tid & 31;
  const int half  = lane >> 4;
  const int l16   = lane & 15;
  const int mBase = wave * 16;
  const int nBase = blockIdx.x * 64;
  const int kw0   = blockIdx.y * chunk;
  const int T     = chunk / KB;

  v8f acc0 = {}, acc1 = {}, acc2 = {}, acc3 = {};

  stage_tiles(A, W, lA[0], lB[0], nBase, K, kw0, tid);

  for (int it = 0; it < T; ++it) {
    async_wait_all();
    __syncthreads();                                   // staged panel visible
    if (it + 1 < T)
      stage_tiles(A, W, lA[(it + 1) & 1], lB[(it + 1) & 1],
                  nBase, K, kw0 + (it + 1) * KB, tid); // fill other buffer async
    const float* tA = &lA[it & 1][(mBase + l16) * LROW + 2 * half];
    const float* tB = &lB[it & 1][l16 * LROW + 2 * half];
#pragma unroll
    for (int kk = 0; kk < KB; kk += 4) {
      v2f a  = *(const v2f*)(tA + kk);
      v2f b0 = *(const v2f*)(tB + kk);
      v2f b1 = *(const v2f*)(tB + 16 * LROW + kk);
      v2f b2 = *(const v2f*)(tB + 32 * LROW + kk);
      v2f b3 = *(const v2f*)(tB + 48 * LROW + kk);
      acc0 = __builtin_amdgcn_wmma_f32_16x16x4_f32(false, a, false, b0, (short)0, acc0, false, false);
      acc1 = __builtin_amdgcn_wmma_f32_16x16x4_f32(false, a, false, b1, (short)0, acc1, false, false);
      acc2 = __builtin_amdgcn_wmma_f32_16x16x4_f32(false, a, false, b2, (short)0, acc2, false, false);
      acc3 = __builtin_amdgcn_wmma_f32_16x16x4_f32(false, a, false, b3, (short)0, acc3, false, false);
    }
    __syncthreads();                                   // done reading buffer
  }

  float* po = part + (size_t)blockIdx.y * B_SZ * N;
  const int mRow = mBase + 8 * half;
#pragma unroll
  for (int r = 0; r < 8; ++r) {
    const size_t base = (size_t)(mRow + r) * N + nBase + l16;
    po[base]      = acc0[r];
    po[base + 16] = acc1[r];
    po[base + 32] = acc2[r];
    po[base + 48] = acc3[r];
  }
}

// ---------------------------------------------------------------------------
// Fused split-K reduction + bias + ELU + LayerNorm. Deterministic (fixed s
// order). Reuses partial slice 0 as the ELU-activation scratch row.
// ---------------------------------------------------------------------------
__global__ void reduce_bias_elu_ln_kernel(float* __restrict__ part, int S,
                                          const float* __restrict__ bias,
                                          const float* __restrict__ g,
                                          const float* __restrict__ beta,
                                          float* __restrict__ out, int N) {
  __shared__ float sm[9];
  const int b = blockIdx.x;
  const size_t stride = (size_t)B_SZ * N;
  float* row0 = part + (size_t)b * N;
  float ssum = 0.f;
  for (int n = threadIdx.x; n < N; n += 256) {
    float v = bias[n];
    for (int s = 0; s < S; ++s) v += part[(size_t)s * stride + (size_t)b * N + n];
    v = (v > 0.f) ? v : expm1f(v);
    row0[n] = v;                         // each (b,n) owned by one thread
    ssum += v;
  }
  const float mean = block_reduce_sum(ssum, sm) / (float)N;
  float vv = 0.f;
  for (int n = threadIdx.x; n < N; n += 256) {
    float d = row0[n] - mean;
    vv += d * d;
  }
  const float var = block_reduce_sum(vv, sm) / (float)N;
  const float rs  = rsqrtf(var + LN_EPS);
  for (int n = threadIdx.x; n < N; n += 256)
    out[(size_t)b * N + n] = (row0[n] - mean) * rs * g[n] + beta[n];
}

// ---------------------------------------------------------------------------
// Head: scores = [h4 | 0.5*x_clin] @ mtlr_w + mtlr_b; out[b,j] = suffix-sum.
// ---------------------------------------------------------------------------
__global__ void head_kernel(const float* __restrict__ h4,
                            const float* __restrict__ xc,
                            const float* __restrict__ mw,
                            const float* __restrict__ mb,
                            float* __restrict__ out) {
  const int b = blockIdx.x;
  const int j = threadIdx.x;             // 32 threads
  __shared__ float sc[N_BINS];
  if (j < N_BINS) {
    float s = mb[j];
#pragma unroll 4
    for (int i = 0; i < D3; ++i) s += h4[(size_t)b * D3 + i] * mw[(size_t)i * N_BINS + j];
#pragma unroll
    for (int i = 0; i < N_CLIN; ++i)
      s += 0.5f * xc[(size_t)b * N_CLIN + i] * mw[(size_t)(D3 + i) * N_BINS + j];
    sc[j] = s;
  }
  __syncthreads();
  if (j <= N_BINS) {
    float acc = 0.f;
    for (int i = j; i < N_BINS; ++i) acc += sc[i];
    out[(size_t)b * (N_BINS + 1) + j] = acc;
  }
}

// ---------------------------------------------------------------------------
extern "C" void kernel_launch(void* const* d_in, const int* in_sizes, int n_in,
                              void* d_out, int out_size, void* d_ws, size_t ws_size,
                              hipStream_t stream) {
  (void)in_sizes; (void)n_in; (void)out_size; (void)ws_size;

  const float* x_main = (const float*)d_in[0];
  const float* x_clin = (const float*)d_in[1];
  const int*   conn   = (const int*)  d_in[2];
  const float* s1_v   = (const float*)d_in[3];
  const float* s1_b   = (const float*)d_in[4];
  const float* l1_g   = (const float*)d_in[5];
  const float* l1_b   = (const float*)d_in[6];
  const float* w2     = (const float*)d_in[7];
  const float* b2     = (const float*)d_in[8];
  const float* l2_g   = (const float*)d_in[9];
  const float* l2_b   = (const float*)d_in[10];
  const float* w3     = (const float*)d_in[11];
  const float* b3     = (const float*)d_in[12];
  const float* l3_g   = (const float*)d_in[13];
  const float* l3_b   = (const float*)d_in[14];
  const float* w4     = (const float*)d_in[15];
  const float* b4     = (const float*)d_in[16];
  const float* l4_g   = (const float*)d_in[17];
  const float* l4_b   = (const float*)d_in[18];
  const float* mtlr_w = (const float*)d_in[19];
  const float* mtlr_b = (const float*)d_in[20];
  float* out = (float*)d_out;

  const int S2 = 8, S3 = 4, S4 = 4;      // split-K factors (chunks of >=128)

  float* ws   = (float*)d_ws;
  float* xT   = ws;  ws += (size_t)IN_DIM * B_SZ;
  float* h1   = ws;  ws += (size_t)B_SZ * D0;
  float* ln1  = ws;  ws += (size_t)B_SZ * D0;
  float* p2   = ws;  ws += (size_t)S2 * B_SZ * D1;
  float* ln2  = ws;  ws += (size_t)B_SZ * D1;
  float* p3   = ws;  ws += (size_t)S3 * B_SZ * D2;
  float* ln3  = ws;  ws += (size_t)B_SZ * D2;
  float* p4   = ws;  ws += (size_t)S4 * B_SZ * D3;
  float* ln4  = ws;  ws += (size_t)B_SZ * D3;

  // 1) transpose x_main -> xT (coalesced gather columns)
  transpose_kernel<<<dim3(IN_DIM / 32, B_SZ / 32), dim3(32, 8), 0, stream>>>(x_main, xT);
  // 2) sparse layer 1 (+bias+ELU), then LN1
  gather_l1_kernel<<<D0, B_SZ, 0, stream>>>(xT, conn, s1_v, s1_b, h1);
  layernorm_kernel<<<B_SZ, 256, 0, stream>>>(h1, l1_g, l1_b, ln1, D0);
  // 3) Layer 2: split-K WMMA GEMM + fused reduce/bias/ELU/LN
  gemm_wmma_splitk_kernel<<<dim3(D1 / 64, S2), 256, 0, stream>>>(ln1, w2, p2, D0, D1, D0 / S2);
  reduce_bias_elu_ln_kernel<<<B_SZ, 256, 0, stream>>>(p2, S2, b2, l2_g, l2_b, ln2, D1);
  // 4) Layer 3
  gemm_wmma_splitk_kernel<<<dim3(D2 / 64, S3), 256, 0, stream>>>(ln2, w3, p3, D1, D2, D1 / S3);
  reduce_bias_elu_ln_kernel<<<B_SZ, 256, 0, stream>>>(p3, S3, b3, l3_g, l3_b, ln3, D2);
  // 5) Layer 4
  gemm_wmma_splitk_kernel<<<dim3(D3 / 64, S4), 256, 0, stream>>>(ln3, w4, p4, D2, D3, D2 / S4);
  reduce_bias_elu_ln_kernel<<<B_SZ, 256, 0, stream>>>(p4, S4, b4, l4_g, l4_b, ln4, D3);
  // 6) MTLR head + suffix-sum over bins
  head_kernel<<<B_SZ, 32, 0, stream>>>(ln4, x_clin, mtlr_w, mtlr_b, out);
}